// FullyConnectedSymmTensorProduct_85710367359119
// MI455X (gfx1250) — compile-verified
//
#include <hip/hip_runtime.h>
#include <stdint.h>

typedef __attribute__((ext_vector_type(2))) float v2f;
typedef __attribute__((ext_vector_type(8))) float v8f;

#define NPATH   11
#define DTOT    576
#define BT      32          // batch rows per block (2 M-tiles of 16)
#define THREADS 256         // 8 wave32: mtile = wave>>2, ntile = wave&3
#define CG_CAP  16          // max CG nonzeros per (path, k-slice)
#define NSLOT   (NPATH * 5)

// ---------------------------------------------------------------------------
// Init kernel: build real Wigner-3j (e3nn convention) with path weights folded
// in, compacted per (path, k) into {i,j packed, coef} entries in d_ws.
// ---------------------------------------------------------------------------
__device__ double dfact(int n) { double r = 1.0; for (int i = 2; i <= n; ++i) r *= (double)i; return r; }

__device__ double cgcoef(int j1, int m1, int j2, int m2, int j3, int m3) {
  if (m3 != m1 + m2) return 0.0;
  double pref = sqrt((2.0 * j3 + 1.0) * dfact(j3 + j1 - j2) * dfact(j3 - j1 + j2)
                     * dfact(j1 + j2 - j3) / dfact(j1 + j2 + j3 + 1));
  pref *= sqrt(dfact(j3 + m3) * dfact(j3 - m3) * dfact(j1 - m1)
               * dfact(j1 + m1) * dfact(j2 - m2) * dfact(j2 + m2));
  double s = 0.0;
  for (int k = 0; k <= j1 + j2; ++k) {
    int d0 = k, d1 = j1 + j2 - j3 - k, d2 = j1 - m1 - k;
    int d3 = j2 + m2 - k, d4 = j3 - j2 + m1 + k, d5 = j3 - j1 - m2 + k;
    if (d0 < 0 || d1 < 0 || d2 < 0 || d3 < 0 || d4 < 0 || d5 < 0) continue;
    double d = dfact(d0) * dfact(d1) * dfact(d2) * dfact(d3) * dfact(d4) * dfact(d5);
    s += ((k & 1) ? -1.0 : 1.0) / d;
  }
  return pref * s;
}

struct cpx { double re, im; };
__device__ inline cpx cmul(cpx a, cpx b) {
  cpx r; r.re = a.re * b.re - a.im * b.im; r.im = a.re * b.im + a.im * b.re; return r;
}

__device__ void build_q(int l, cpx q[5][5]) {
  for (int a = 0; a < 5; ++a) for (int b = 0; b < 5; ++b) { q[a][b].re = 0.0; q[a][b].im = 0.0; }
  double s2 = 1.0 / sqrt(2.0);
  for (int m = -l; m < 0; ++m) {
    q[l + m][l - m].re = s2;            // 1/sqrt2
    q[l + m][l + m].im = -s2;           // -i/sqrt2
  }
  q[l][l].re = 1.0;
  for (int m = 1; m <= l; ++m) {
    double sg = (m & 1) ? -1.0 : 1.0;
    q[l + m][l + m].re = sg * s2;       // (-1)^m/sqrt2
    q[l + m][l - m].im = sg * s2;       // i(-1)^m/sqrt2
  }
  cpx ph;                               // (-i)^l
  if (l == 0)      { ph.re = 1.0;  ph.im = 0.0; }
  else if (l == 1) { ph.re = 0.0;  ph.im = -1.0; }
  else             { ph.re = -1.0; ph.im = 0.0; }
  for (int a = 0; a < 5; ++a) for (int b = 0; b < 5; ++b) q[a][b] = cmul(ph, q[a][b]);
}

__global__ void cg_init_kernel(int* cnt, uint2* tab) {
  if (threadIdx.x != 0 || blockIdx.x != 0) return;
  const int L1[NPATH] = {0, 0, 0, 1, 1, 1, 1, 2, 2, 2, 2};
  const int L2[NPATH] = {0, 1, 2, 0, 1, 1, 2, 0, 1, 2, 2};
  const int L3[NPATH] = {0, 1, 2, 1, 0, 2, 1, 2, 1, 0, 2};
  const int NPL[3]    = {3, 4, 4};    // #paths feeding each l3 (fan-in / 4096)
  for (int s = 0; s < NSLOT; ++s) cnt[s] = 0;

  for (int p = 0; p < NPATH; ++p) {
    int l1 = L1[p], l2 = L2[p], l3 = L3[p];
    int n1 = 2 * l1 + 1, n2 = 2 * l2 + 1, n3 = 2 * l3 + 1;
    cpx q1[5][5], q2[5][5], q3[5][5];
    build_q(l1, q1); build_q(l2, q2); build_q(l3, q3);
    double pw = sqrt((double)n3 / (4096.0 * NPL[l3]));
    double scale = pw / sqrt((double)n3);   // path weight * 1/sqrt(2l3+1)
    for (int k = 0; k < n3; ++k) {
      int c = 0;
      for (int i = 0; i < n1; ++i) {
        for (int j = 0; j < n2; ++j) {
          cpx acc; acc.re = 0.0; acc.im = 0.0;
          for (int a = 0; a < n1; ++a) {
            int m1 = a - l1;
            for (int b = 0; b < n2; ++b) {
              int m2 = b - l2;
              int m3 = m1 + m2;
              if (m3 < -l3 || m3 > l3) continue;
              double Cv = cgcoef(l1, m1, l2, m2, l3, m3);
              if (Cv == 0.0) continue;
              int nn = l3 + m3;
              cpx t = cmul(q1[a][i], q2[b][j]);
              cpx q3c; q3c.re = q3[nn][k].re; q3c.im = -q3[nn][k].im;
              t = cmul(t, q3c);
              acc.re += t.re * Cv; acc.im += t.im * Cv;
            }
          }
          double v = acc.re * scale;
          if ((v > 1e-10 || v < -1e-10) && c < CG_CAP) {
            uint2 e;
            e.x = (uint32_t)(i | (j << 8));
            e.y = __float_as_uint((float)v);
            tab[(p * 5 + k) * CG_CAP + c] = e;
            ++c;
          }
        }
      }
      cnt[p * 5 + k] = c;
    }
  }
}

// ---------------------------------------------------------------------------
// Main kernel: block = 32 batch rows, 8 waves = 2 M-tiles x 4 N-tiles.
// Per path: stage full y-block in LDS (skipped when l2 repeats); loop u, and
// per u stage a tiny x slice; per 16-wide uv chunk build z + symmetrized-W
// tiles in LDS, then 4 K-steps x (2l3+1) k-slices of f32 WMMA per wave.
// ---------------------------------------------------------------------------
template <int P, int L1, int L2, int L3, int ABASE, bool RESTAGE_Y>
__device__ __forceinline__ void run_path(
    const float* __restrict__ x, const float* __restrict__ y,
    const float* __restrict__ W,
    float* xu, float* ys, float* zs, float* wsm,
    const int* s_cnt, const uint2* s_tab,
    v8f (&acc)[9], int bBase, int tid, int lane, int mtile, int ntile)
{
  constexpr int N1 = 2 * L1 + 1, N2 = 2 * L2 + 1, N3 = 2 * L3 + 1;
  constexpr int P2 = 64 * N2 + 1;   // LDS pitch per batch row (odd -> bank spread)
  constexpr int XO = (L1 == 0 ? 0 : (L1 == 1 ? 64 : 256));
  constexpr int YO = (L2 == 0 ? 0 : (L2 == 1 ? 64 : 256));

  if constexpr (RESTAGE_Y) {
    __syncthreads();  // previous path fully consumed ys (and initial s_tab copy)
    for (int idx = tid; idx < BT * 64 * N2; idx += THREADS) {
      int b = idx / (64 * N2);
      int r = idx - b * (64 * N2);
      ys[b * P2 + r] = y[(size_t)(bBase + b) * DTOT + YO + r];
    }
    __syncthreads();
  }

  const int cpos = tid & 15;   // column within uv chunk
  const int b0   = tid >> 4;   // 0..15 -> handles rows b0 and b0+16
  const float* Wp = W + (size_t)P * (64 * 64 * 64);

  for (int u = 0; u < 64; ++u) {
    // Stage x slice for this u: xu[b*8 + i], b = 0..31, i < N1.
    // Safe without a leading barrier: all reads of the previous slice finished
    // before the previous chunk's mid-barrier; visibility is provided by the
    // chunk-top barrier below.
    for (int idx = tid; idx < BT * N1; idx += THREADS) {
      int b = idx / N1;
      int i = idx - b * N1;
      xu[b * 8 + i] = x[(size_t)(bBase + b) * DTOT + XO + u * N1 + i];
    }

    for (int v0 = 0; v0 < 64; v0 += 16) {
      __syncthreads();         // previous chunk's zs/wsm consumed; xu visible

      // ---- z tile: z[b, (u,v0+cpos), k] via sparse CG lists ----
      {
        const int v = v0 + cpos;
        #pragma unroll
        for (int half = 0; half < 2; ++half) {
          const int b = b0 + half * 16;
          const float* xr = xu + b * 8;
          const float* yr = ys + b * P2 + v * N2;
          #pragma unroll
          for (int ks = 0; ks < N3; ++ks) {
            float zk = 0.f;
            const int slot = P * 5 + ks;
            const int n = s_cnt[slot];
            const uint2* e = s_tab + slot * CG_CAP;
            for (int t = 0; t < n; ++t) {
              uint2 en = e[t];
              int i = (int)(en.x & 255u), j = (int)((en.x >> 8) & 255u);
              zk = fmaf(__uint_as_float(en.y), xr[i] * yr[j], zk);
            }
            zs[ks * (BT * 17) + b * 17 + cpos] = zk;
          }
        }
      }
      // ---- symmetrized W tile: wsm[c][w], pitch 80 ----
      {
        const int w  = tid & 63;
        const int cb = tid >> 6;  // 0..3
        #pragma unroll
        for (int q = 0; q < 4; ++q) {
          const int c = cb * 4 + q;
          const int v = v0 + c;
          float a0 = Wp[((size_t)u * 64 + v) * 64 + w];
          float a1 = Wp[((size_t)v * 64 + u) * 64 + w];
          wsm[c * 80 + w] = 0.5f * (a0 + a1);
        }
        if (u * 64 + v0 + 16 < 4096) {
          const int nxt = u * 64 + v0 + 16;
          const int un = nxt >> 6, vn = nxt & 63;
          __builtin_prefetch(&Wp[((size_t)un * 64 + vn) * 64 + (tid & 63)], 0, 0);
        }
      }
      __syncthreads();

      // ---- WMMA: D(16b x 16w) += A(16b x 4uv) * B(4uv x 16w), 4 K-steps ----
      const int n     = lane & 15;
      const int khalf = lane >> 4;
      const int wcol  = ntile * 16 + n;
      const int arow  = mtile * 16 + n;
      #pragma unroll
      for (int kk = 0; kk < 16; kk += 4) {
        v2f bf;
        bf.x = wsm[(kk + 2 * khalf) * 80 + wcol];
        bf.y = wsm[(kk + 2 * khalf + 1) * 80 + wcol];
        #pragma unroll
        for (int ks = 0; ks < N3; ++ks) {
          const float* zrow = zs + ks * (BT * 17) + arow * 17 + kk + 2 * khalf;
          v2f af;
          af.x = zrow[0];
          af.y = zrow[1];
          acc[ABASE + ks] = __builtin_amdgcn_wmma_f32_16x16x4_f32(
              false, af, false, bf, (short)0, acc[ABASE + ks], false, false);
        }
      }
    }
  }
}

__global__ void __launch_bounds__(THREADS)
tp_kernel(const float* __restrict__ x, const float* __restrict__ y,
          const float* __restrict__ W, float* __restrict__ out,
          const int* __restrict__ g_cnt, const uint2* __restrict__ g_tab)
{
  __shared__ float ys[BT * 321];        // 41,088 B
  __shared__ float zs[5 * BT * 17];     // 10,880 B
  __shared__ float wsm[16 * 80];        //  5,120 B
  __shared__ float xu[BT * 8];          //  1,024 B
  __shared__ int   s_cnt[NSLOT];        //    220 B
  __shared__ uint2 s_tab[NSLOT * CG_CAP]; // 7,040 B   (total ~65.4 KB)

  const int tid   = threadIdx.x;
  const int lane  = tid & 31;
  const int wave  = tid >> 5;
  const int ntile = wave & 3;           // w-tile (0..3)
  const int mtile = wave >> 2;          // batch half-tile (0..1)
  const int bBase = blockIdx.x * BT;

  for (int i = tid; i < NSLOT; i += THREADS) s_cnt[i] = g_cnt[i];
  for (int i = tid; i < NSLOT * CG_CAP; i += THREADS) s_tab[i] = g_tab[i];

  v8f acc[9];
  #pragma unroll
  for (int i = 0; i < 9; ++i) acc[i] = {};

  // INSTR order; acc bases: l3=0 -> acc[0], l3=1 -> acc[1..3], l3=2 -> acc[4..8]
  // RESTAGE_Y=false when the previous path already staged the same l2 block.
  run_path< 0, 0, 0, 0, 0, true >(x, y, W, xu, ys, zs, wsm, s_cnt, s_tab, acc, bBase, tid, lane, mtile, ntile);
  run_path< 1, 0, 1, 1, 1, true >(x, y, W, xu, ys, zs, wsm, s_cnt, s_tab, acc, bBase, tid, lane, mtile, ntile);
  run_path< 2, 0, 2, 2, 4, true >(x, y, W, xu, ys, zs, wsm, s_cnt, s_tab, acc, bBase, tid, lane, mtile, ntile);
  run_path< 3, 1, 0, 1, 1, true >(x, y, W, xu, ys, zs, wsm, s_cnt, s_tab, acc, bBase, tid, lane, mtile, ntile);
  run_path< 4, 1, 1, 0, 0, true >(x, y, W, xu, ys, zs, wsm, s_cnt, s_tab, acc, bBase, tid, lane, mtile, ntile);
  run_path< 5, 1, 1, 2, 4, false>(x, y, W, xu, ys, zs, wsm, s_cnt, s_tab, acc, bBase, tid, lane, mtile, ntile);
  run_path< 6, 1, 2, 1, 1, true >(x, y, W, xu, ys, zs, wsm, s_cnt, s_tab, acc, bBase, tid, lane, mtile, ntile);
  run_path< 7, 2, 0, 2, 4, true >(x, y, W, xu, ys, zs, wsm, s_cnt, s_tab, acc, bBase, tid, lane, mtile, ntile);
  run_path< 8, 2, 1, 1, 1, true >(x, y, W, xu, ys, zs, wsm, s_cnt, s_tab, acc, bBase, tid, lane, mtile, ntile);
  run_path< 9, 2, 2, 0, 0, true >(x, y, W, xu, ys, zs, wsm, s_cnt, s_tab, acc, bBase, tid, lane, mtile, ntile);
  run_path<10, 2, 2, 2, 4, false>(x, y, W, xu, ys, zs, wsm, s_cnt, s_tab, acc, bBase, tid, lane, mtile, ntile);

  // Epilogue: C/D layout -> lane n = w column, vgpr r (+8 for upper lane half)
  // = batch row within this wave's 16-row M-tile.
  const int n     = lane & 15;
  const int khalf = lane >> 4;
  const int wc    = ntile * 16 + n;
  #pragma unroll
  for (int r = 0; r < 8; ++r) {
    const int b = bBase + mtile * 16 + r + khalf * 8;
    float* ob = out + (size_t)b * DTOT;
    ob[wc] = acc[0][r];                                    // l3=0 block (w stride 1)
    #pragma unroll
    for (int ks = 0; ks < 3; ++ks) ob[64 + wc * 3 + ks] = acc[1 + ks][r];   // l3=1
    #pragma unroll
    for (int ks = 0; ks < 5; ++ks) ob[256 + wc * 5 + ks] = acc[4 + ks][r];  // l3=2
  }
}

// ---------------------------------------------------------------------------
extern "C" void kernel_launch(void* const* d_in, const int* in_sizes, int n_in,
                              void* d_out, int out_size, void* d_ws, size_t ws_size,
                              hipStream_t stream) {
  const float* x = (const float*)d_in[0];
  const float* y = (const float*)d_in[1];
  const float* W = (const float*)d_in[2];
  float* out = (float*)d_out;

  int*   cnt = (int*)d_ws;
  uint2* tab = (uint2*)((char*)d_ws + 256);

  cg_init_kernel<<<1, 1, 0, stream>>>(cnt, tab);
  tp_kernel<<<8192 / BT, THREADS, 0, stream>>>(x, y, W, out, cnt, tab);
}